// GNN_node_Virtualnode_4655744549604
// MI455X (gfx1250) — compile-verified
//
#include <hip/hip_runtime.h>

// ---------------------------------------------------------------------------
// Types for CDNA5 WMMA f32 16x16x4
// ---------------------------------------------------------------------------
typedef __attribute__((ext_vector_type(2))) float v2f;
typedef __attribute__((ext_vector_type(8))) float v8f;

// ---------------------------------------------------------------------------
// Generic GEMM + bias:  C[M,Nc] = A[M,K] @ B[K,Nc] + bias[Nc]
// One wave computes a 16 x (16*TN) output tile with V_WMMA_F32_16X16X4_F32,
// reusing one A fragment across TN accumulators (4x less A traffic).
// A-layout (16x4 f32): lanes 0-15 hold M=0..15, VGPR0=K0,VGPR1=K1;
//                      lanes 16-31 hold K2,K3.
// B-layout (4x16 f32): mirrored: lanes hold N=0..15, VGPRs hold K pairs.
// C-layout (16x16 f32, 8 VGPRs): VGPR r = row (r + 8*half), col = lane&15.
// Early exit is wave-uniform, so EXEC is all-ones at every WMMA.
// ---------------------------------------------------------------------------
template <int K, int TN>
__global__ __launch_bounds__(256) void wmma_gemm_bias_kernel(
    const float* __restrict__ A, const float* __restrict__ B,
    const float* __restrict__ bias, float* __restrict__ C, int M, int Nc) {
  int wave = (blockIdx.x * blockDim.x + threadIdx.x) >> 5;
  int lane = threadIdx.x & 31;
  int tilesM = M >> 4;
  int groupsN = Nc / (16 * TN);
  if (wave >= tilesM * groupsN) return;  // wave-uniform
  int tm = wave % tilesM;
  int tg = wave / tilesM;
  int m0 = tm << 4;
  int n0 = tg * (16 * TN);
  int hi = lane >> 4;   // which half of the wave
  int l16 = lane & 15;

  const float* Arow = A + (size_t)(m0 + l16) * K;
  const float* Bcol = B + n0 + l16;

  v8f acc[TN];
#pragma unroll
  for (int t = 0; t < TN; ++t) acc[t] = (v8f){0.f, 0.f, 0.f, 0.f, 0.f, 0.f, 0.f, 0.f};

#pragma unroll 4
  for (int k = 0; k < K; k += 4) {
    int ka = k + 2 * hi;
    v2f a;
    a.x = Arow[ka];
    a.y = Arow[ka + 1];
    const float* b0 = Bcol + (size_t)ka * Nc;
    const float* b1 = Bcol + (size_t)(ka + 1) * Nc;
#pragma unroll
    for (int t = 0; t < TN; ++t) {
      v2f b;
      b.x = b0[t * 16];
      b.y = b1[t * 16];
      acc[t] = __builtin_amdgcn_wmma_f32_16x16x4_f32(
          /*neg_a=*/false, a, /*neg_b=*/false, b,
          /*c_mod=*/(short)0, acc[t], /*reuse_a=*/false, /*reuse_b=*/false);
    }
  }
#pragma unroll
  for (int t = 0; t < TN; ++t) {
    float bv = bias ? bias[n0 + t * 16 + l16] : 0.f;
#pragma unroll
    for (int r = 0; r < 8; ++r) {
      C[(size_t)(m0 + r + 8 * hi) * Nc + n0 + t * 16 + l16] = acc[t][r] + bv;
    }
  }
}

// ---------------------------------------------------------------------------
// h_in = h + vn[batch[n]] ; agg = 0     (i over N*128)
// ---------------------------------------------------------------------------
__global__ void add_vn_zero_agg_kernel(const float* __restrict__ h,
                                       const float* __restrict__ vn,
                                       const int* __restrict__ batch,
                                       float* __restrict__ h_in,
                                       float* __restrict__ agg, int total) {
  int i = blockIdx.x * blockDim.x + threadIdx.x;
  if (i >= total) return;
  int n = i >> 7;
  int c = i & 127;
  h_in[i] = h[i] + vn[(size_t)batch[n] * 128 + c];
  agg[i] = 0.f;
}

// ---------------------------------------------------------------------------
// Edge message + scatter:
//   emb = edge_attr[e] @ Wl + bl        (8 -> 128, Wl staged in LDS)
//   msg = relu(h_in[src[e]] + emb)
//   atomicAdd(agg[dst[e]], msg)
// One wave per edge iteration; each lane owns 4 consecutive channels.
// ---------------------------------------------------------------------------
__global__ __launch_bounds__(256) void edge_msg_scatter_kernel(
    const float* __restrict__ edge_attr, const int* __restrict__ edge_index,
    const float* __restrict__ h_in, const float* __restrict__ Wl,
    const float* __restrict__ bl, float* __restrict__ agg, int E) {
  __shared__ float sW[8 * 128];
  __shared__ float sB[128];
  for (int i = threadIdx.x; i < 8 * 128; i += blockDim.x) sW[i] = Wl[i];
  for (int i = threadIdx.x; i < 128; i += blockDim.x) sB[i] = bl[i];
  __syncthreads();

  int lane = threadIdx.x & 31;
  int wave = (blockIdx.x * blockDim.x + threadIdx.x) >> 5;
  int nwaves = (gridDim.x * blockDim.x) >> 5;
  int c0 = lane << 2;

  for (int e = wave; e < E; e += nwaves) {
    int s = edge_index[e];
    int d = edge_index[E + e];
    const float4 ea0 = *(const float4*)(edge_attr + (size_t)e * 8);
    const float4 ea1 = *(const float4*)(edge_attr + (size_t)e * 8 + 4);
    float4 hv = *(const float4*)(h_in + (size_t)s * 128 + c0);
    float m[4];
#pragma unroll
    for (int c = 0; c < 4; ++c) {
      int col = c0 + c;
      float emb = sB[col];
      emb += ea0.x * sW[0 * 128 + col];
      emb += ea0.y * sW[1 * 128 + col];
      emb += ea0.z * sW[2 * 128 + col];
      emb += ea0.w * sW[3 * 128 + col];
      emb += ea1.x * sW[4 * 128 + col];
      emb += ea1.y * sW[5 * 128 + col];
      emb += ea1.z * sW[6 * 128 + col];
      emb += ea1.w * sW[7 * 128 + col];
      float hval = (c == 0) ? hv.x : (c == 1) ? hv.y : (c == 2) ? hv.z : hv.w;
      float v = hval + emb;
      m[c] = v > 0.f ? v : 0.f;
    }
    float* base = agg + (size_t)d * 128 + c0;
    atomicAdd(base + 0, m[0]);
    atomicAdd(base + 1, m[1]);
    atomicAdd(base + 2, m[2]);
    atomicAdd(base + 3, m[3]);
  }
}

// z_in = (1+eps[l]) * h_in + agg      (in place over agg)
__global__ void combine_kernel(const float* __restrict__ h_in,
                               float* __restrict__ agg,
                               const float* __restrict__ eps, int l,
                               int total) {
  int i = blockIdx.x * blockDim.x + threadIdx.x;
  if (i >= total) return;
  agg[i] = (1.f + eps[l]) * h_in[i] + agg[i];
}

// Per-channel sum & sum-of-squares. gridDim*blockDim must be a multiple of C
// so each thread's strided elements all share one channel.
__global__ void bn_stats_kernel(const float* __restrict__ X,
                                float* __restrict__ sums, int total, int C) {
  int idx = blockIdx.x * blockDim.x + threadIdx.x;
  int stride = gridDim.x * blockDim.x;
  float s = 0.f, q = 0.f;
  for (int i = idx; i < total; i += stride) {
    float x = X[i];
    s += x;
    q += x * x;
  }
  int c = idx % C;
  atomicAdd(sums + c, s);
  atomicAdd(sums + C + c, q);
}

// y = (x - mean)*rsqrt(var+eps)*g + b, optional relu.  (Y may alias X)
__global__ void bn_apply_kernel(const float* __restrict__ X,
                                const float* __restrict__ sums,
                                const float* __restrict__ g,
                                const float* __restrict__ b,
                                float* __restrict__ Y, int M, int C,
                                int relu) {
  int i = blockIdx.x * blockDim.x + threadIdx.x;
  long long total = (long long)M * C;
  if (i >= total) return;
  int c = i % C;
  float invM = 1.f / (float)M;
  float mean = sums[c] * invM;
  float var = sums[C + c] * invM - mean * mean;
  float rstd = rsqrtf(var + 1e-5f);
  float y = (X[i] - mean) * rstd * g[c] + b[c];
  if (relu) y = y > 0.f ? y : 0.f;
  Y[i] = y;
}

__global__ void zero_f32_kernel(float* __restrict__ p, int count) {
  int i = blockIdx.x * blockDim.x + threadIdx.x;
  if (i < count) p[i] = 0.f;
}

// vn[g,c] = vn_emb[c]   (i over G*128)
__global__ void vn_from_emb_kernel(float* __restrict__ vn,
                                   const float* __restrict__ emb, int total) {
  int i = blockIdx.x * blockDim.x + threadIdx.x;
  if (i >= total) return;
  vn[i] = emb[i & 127];
}

__global__ void copy_f32_kernel(float* __restrict__ dst,
                                const float* __restrict__ src, int count) {
  int i = blockIdx.x * blockDim.x + threadIdx.x;
  if (i < count) dst[i] = src[i];
}

// vtmp[batch[n]] += h_in[n]   (i over N*128)
__global__ void vn_pool_kernel(const float* __restrict__ h_in,
                               const int* __restrict__ batch,
                               float* __restrict__ vtmp, int total) {
  int i = blockIdx.x * blockDim.x + threadIdx.x;
  if (i >= total) return;
  int n = i >> 7;
  int c = i & 127;
  atomicAdd(vtmp + (size_t)batch[n] * 128 + c, h_in[i]);
}

// ---------------------------------------------------------------------------
// Launcher
// ---------------------------------------------------------------------------
extern "C" void kernel_launch(void* const* d_in, const int* in_sizes, int n_in,
                              void* d_out, int out_size, void* d_ws,
                              size_t ws_size, hipStream_t stream) {
  (void)in_sizes; (void)n_in; (void)out_size; (void)ws_size;
  const int N = 50000, E = 400000, G = 256, D = 128, L = 3;
  const int D2 = 2 * D;  // 256

  const float* x         = (const float*)d_in[0];
  const float* edge_attr = (const float*)d_in[1];
  const int*   edge_idx  = (const int*)d_in[2];
  const int*   batch     = (const int*)d_in[3];
  const float* node_W    = (const float*)d_in[4];
  const float* node_b    = (const float*)d_in[5];
  const float* vn_emb    = (const float*)d_in[6];
  const float* edge_W    = (const float*)d_in[7];
  const float* edge_b    = (const float*)d_in[8];
  const float* eps       = (const float*)d_in[9];
  const float* gin_W1    = (const float*)d_in[10];
  const float* gin_b1    = (const float*)d_in[11];
  const float* gin_bn1_g = (const float*)d_in[12];
  const float* gin_bn1_b = (const float*)d_in[13];
  const float* gin_W2    = (const float*)d_in[14];
  const float* gin_b2    = (const float*)d_in[15];
  const float* bn_g      = (const float*)d_in[16];
  const float* bn_b      = (const float*)d_in[17];
  const float* vmlp_W1   = (const float*)d_in[18];
  const float* vmlp_b1   = (const float*)d_in[19];
  const float* vmlp_bn1_g= (const float*)d_in[20];
  const float* vmlp_bn1_b= (const float*)d_in[21];
  const float* vmlp_W2   = (const float*)d_in[22];
  const float* vmlp_b2   = (const float*)d_in[23];
  const float* vmlp_bn2_g= (const float*)d_in[24];
  const float* vmlp_bn2_b= (const float*)d_in[25];

  // Workspace layout (floats). h/h_in/agg: N*D each; z1: N*2D; small tails.
  float* ws   = (float*)d_ws;
  float* h    = ws;                       // 6,400,000
  float* h_in = ws + 6400000;             // 6,400,000
  float* agg  = ws + 12800000;            // 6,400,000  (also reused as z_in)
  float* z1   = ws + 19200000;            // 12,800,000
  float* vn   = ws + 32000000;            // G*D   = 32768
  float* vtmp = vn + G * D;               // G*D   = 32768
  float* v1   = vtmp + G * D;             // G*2D  = 65536
  float* sums = v1 + G * D2;              // 512
  float* out  = (float*)d_out;

  const int ND = N * D;       // 6.4M
  const int ND2 = N * D2;     // 12.8M
  const int TB = 256;
  auto blk = [](int total) { return (total + 255) / 256; };
  // GEMM grids: one wave per 16x(16*TN) tile, 8 waves per block, TN=4.
  auto gblk = [](int M, int Nc) { return ((M / 16) * (Nc / 64) + 7) / 8; };

  // ---- node encoder: h = x @ node_W + node_b   [N,16]x[16,128]
  wmma_gemm_bias_kernel<16, 4><<<gblk(N, D), TB, 0, stream>>>(
      x, node_W, node_b, h, N, D);
  // ---- vn = broadcast(vn_emb)
  vn_from_emb_kernel<<<blk(G * D), TB, 0, stream>>>(vn, vn_emb, G * D);

  for (int l = 0; l < L; ++l) {
    // h_in = h + vn[batch]; agg = 0
    add_vn_zero_agg_kernel<<<blk(ND), TB, 0, stream>>>(h, vn, batch, h_in, agg,
                                                       ND);
    // edge messages -> agg
    edge_msg_scatter_kernel<<<4096, TB, 0, stream>>>(
        edge_attr, edge_idx, h_in, edge_W + (size_t)l * 8 * D,
        edge_b + (size_t)l * D, agg, E);
    // z_in = (1+eps)*h_in + agg   (in place in agg)
    combine_kernel<<<blk(ND), TB, 0, stream>>>(h_in, agg, eps, l, ND);

    // z1 = z_in @ gin_W1 + b1   [N,128]x[128,256]
    wmma_gemm_bias_kernel<128, 4><<<gblk(N, D2), TB, 0, stream>>>(
        agg, gin_W1 + (size_t)l * D * D2, gin_b1 + (size_t)l * D2, z1, N, D2);
    // BN1 + relu (in place)
    zero_f32_kernel<<<1, 512, 0, stream>>>(sums, 2 * D2);
    bn_stats_kernel<<<512, TB, 0, stream>>>(z1, sums, ND2, D2);
    bn_apply_kernel<<<blk(ND2), TB, 0, stream>>>(
        z1, sums, gin_bn1_g + (size_t)l * D2, gin_bn1_b + (size_t)l * D2, z1,
        N, D2, 1);

    // z2 = z1 @ gin_W2 + b2   [N,256]x[256,128]  (write into h; old h is dead)
    wmma_gemm_bias_kernel<256, 4><<<gblk(N, D), TB, 0, stream>>>(
        z1, gin_W2 + (size_t)l * D2 * D, gin_b2 + (size_t)l * D, h, N, D);
    // outer BN (+relu except last layer); last layer writes d_out
    zero_f32_kernel<<<1, 512, 0, stream>>>(sums, 2 * D);
    bn_stats_kernel<<<512, TB, 0, stream>>>(h, sums, ND, D);
    float* hout = (l == L - 1) ? out : h;
    bn_apply_kernel<<<blk(ND), TB, 0, stream>>>(h, sums, bn_g + (size_t)l * D,
                                                bn_b + (size_t)l * D, hout, N,
                                                D, (l < L - 1) ? 1 : 0);

    if (l < L - 1) {
      // vtmp = vn + segment_sum(h_in, batch)
      copy_f32_kernel<<<blk(G * D), TB, 0, stream>>>(vtmp, vn, G * D);
      vn_pool_kernel<<<blk(ND), TB, 0, stream>>>(h_in, batch, vtmp, ND);
      // v1 = vtmp @ vmlp_W1 + b1   [256,128]x[128,256]
      wmma_gemm_bias_kernel<128, 4><<<gblk(G, D2), TB, 0, stream>>>(
          vtmp, vmlp_W1 + (size_t)l * D * D2, vmlp_b1 + (size_t)l * D2, v1, G,
          D2);
      zero_f32_kernel<<<1, 512, 0, stream>>>(sums, 2 * D2);
      bn_stats_kernel<<<2, TB, 0, stream>>>(v1, sums, G * D2, D2);
      bn_apply_kernel<<<blk(G * D2), TB, 0, stream>>>(
          v1, sums, vmlp_bn1_g + (size_t)l * D2, vmlp_bn1_b + (size_t)l * D2,
          v1, G, D2, 1);
      // vn_new = bn(relu) of v1 @ vmlp_W2 + b2  (write into vn; old vn dead)
      wmma_gemm_bias_kernel<256, 4><<<gblk(G, D), TB, 0, stream>>>(
          v1, vmlp_W2 + (size_t)l * D2 * D, vmlp_b2 + (size_t)l * D, vn, G, D);
      zero_f32_kernel<<<1, 512, 0, stream>>>(sums, 2 * D);
      bn_stats_kernel<<<2, TB, 0, stream>>>(vn, sums, G * D, D);
      bn_apply_kernel<<<blk(G * D), TB, 0, stream>>>(
          vn, sums, vmlp_bn2_g + (size_t)l * D, vmlp_bn2_b + (size_t)l * D, vn,
          G, D, 1);
    }
  }
}